// ProteinEmbeddingPooling_3813930958921
// MI455X (gfx1250) — compile-verified
//
#include <hip/hip_runtime.h>
#include <hip/hip_bf16.h>

// Problem constants (from reference setup_inputs)
#define BB   32
#define SS   1024
#define DD   1280
#define DOUT 1280
#define LN_EPS    1e-5f
#define MASK_FILL -1e9f

typedef __attribute__((ext_vector_type(2))) float v2f;
typedef __attribute__((ext_vector_type(8))) float v8f;

// ---------------------------------------------------------------------------
// WMMA fp32 GEMM: C[32 x N] = scale * (A[32 x K] * B) + bias
//   MODE 0: B[k][n] = W[k*ldb + n]        (A @ W)
//   MODE 1: B[k][n] = W[n*ldb + k]        (A @ W^T)
// One wave (32 threads) per 16x16 output tile. grid = (N/16, 2).
// A fragment (f32 16x4): lane L holds M = L%16, K = 2*(L/16) + {vgpr0, vgpr1}
// B fragment (f32 4x16): lane L holds N = L%16, K = 2*(L/16) + {vgpr0, vgpr1}
// C/D (f32 16x16): lane L, vgpr r holds M = 8*(L/16) + r, N = L%16
// ---------------------------------------------------------------------------
template <int MODE>
__global__ void gemm32_wmma_kernel(const float* __restrict__ A, long lda_row,
                                   const float* __restrict__ W, int ldb,
                                   const float* __restrict__ bias,
                                   float* __restrict__ C, int ldc,
                                   int K, float scale) {
    const int lane = threadIdx.x;        // 0..31
    const int row  = lane & 15;          // M (for A) / N (for B)
    const int half = lane >> 4;          // selects K pair
    const int m0   = blockIdx.y * 16;
    const int n0   = blockIdx.x * 16;

    const float* arow = A + (long)(m0 + row) * lda_row;

    v8f acc0 = {}, acc1 = {}, acc2 = {}, acc3 = {};

    for (int k0 = 0; k0 < K; k0 += 16) {
        v8f* accs[4] = {&acc0, &acc1, &acc2, &acc3};
#pragma unroll
        for (int u = 0; u < 4; ++u) {
            const int k = k0 + u * 4 + half * 2;   // this lane's K base
            v2f a;
            a.x = arow[k];
            a.y = arow[k + 1];
            v2f b;
            if (MODE == 0) {
                b.x = W[(long)k * ldb + n0 + row];
                b.y = W[(long)(k + 1) * ldb + n0 + row];
            } else {
                const float* wp = W + (long)(n0 + row) * ldb + k;
                b.x = wp[0];
                b.y = wp[1];
            }
            *accs[u] = __builtin_amdgcn_wmma_f32_16x16x4_f32(
                false, a, false, b, (short)0, *accs[u], false, false);
        }
    }

    v8f c = (acc0 + acc1) + (acc2 + acc3);
    const int n = n0 + row;
    const float bval = bias ? bias[n] : 0.0f;
#pragma unroll
    for (int r = 0; r < 8; ++r) {
        const int m = m0 + half * 8 + r;
        C[(long)m * ldc + n] = scale * c[r] + bval;
    }
}

// ---------------------------------------------------------------------------
// qbk[b] = scale * dot(q0[b,:], bk)
// ---------------------------------------------------------------------------
__global__ void qdot_kernel(const float* __restrict__ q0,
                            const float* __restrict__ bk,
                            float* __restrict__ qbk, float scale) {
    __shared__ float red[256];
    const int b = blockIdx.x, tid = threadIdx.x;
    float s = 0.0f;
#pragma unroll
    for (int i = 0; i < 5; ++i) {
        const int e = tid + 256 * i;
        s += q0[(long)b * DD + e] * bk[e];
    }
    red[tid] = s; __syncthreads();
    for (int st = 128; st > 0; st >>= 1) {
        if (tid < st) red[tid] += red[tid + st];
        __syncthreads();
    }
    if (tid == 0) qbk[b] = scale * red[0];
}

// ---------------------------------------------------------------------------
// Pass 1 (HBM-bound): scores[b,s] = dot(emb[b,s,:], kq[b,:]) + qbk[b], masked.
// One wave per row, float4 loads, shuffle reduction. kq is pre-scaled.
// ---------------------------------------------------------------------------
__global__ void scores_kernel(const float* __restrict__ emb,
                              const float* __restrict__ kq,
                              const float* __restrict__ qbk,
                              const int* __restrict__ mask,
                              float* __restrict__ scores) {
    const int wave = threadIdx.x >> 5;
    const int lane = threadIdx.x & 31;
    const long row = (long)blockIdx.x * 8 + wave;   // 0..32767
    const int b = (int)(row >> 10);

    const float4* e4 = (const float4*)(emb + row * DD);
    const float4* k4 = (const float4*)(kq + (long)b * DD);

    float acc = 0.0f;
#pragma unroll
    for (int i = 0; i < 10; ++i) {                  // 1280 = 10 * 32 * 4
        const float4 ev = e4[lane + 32 * i];
        const float4 kv = k4[lane + 32 * i];
        acc += ev.x * kv.x + ev.y * kv.y + ev.z * kv.z + ev.w * kv.w;
    }
#pragma unroll
    for (int off = 16; off > 0; off >>= 1)
        acc += __shfl_xor(acc, off, 32);

    if (lane == 0) {
        const float v = acc + qbk[b];
        scores[row] = (mask[row] == 0) ? MASK_FILL : v;
    }
}

// ---------------------------------------------------------------------------
// Softmax over S=1024 per batch. grid=32, block=256 (4 elems/thread).
// ---------------------------------------------------------------------------
__global__ void softmax_kernel(const float* __restrict__ scores,
                               float* __restrict__ weights) {
    __shared__ float red[256];
    const int b = blockIdx.x, tid = threadIdx.x;
    const float* sc = scores + (long)b * SS;

    float v[4];
    float m = -3.4e38f;
#pragma unroll
    for (int i = 0; i < 4; ++i) {
        v[i] = sc[tid + 256 * i];
        m = fmaxf(m, v[i]);
    }
    red[tid] = m; __syncthreads();
    for (int st = 128; st > 0; st >>= 1) {
        if (tid < st) red[tid] = fmaxf(red[tid], red[tid + st]);
        __syncthreads();
    }
    m = red[0]; __syncthreads();

    float e[4], l = 0.0f;
#pragma unroll
    for (int i = 0; i < 4; ++i) {
        e[i] = __expf(v[i] - m);
        l += e[i];
    }
    red[tid] = l; __syncthreads();
    for (int st = 128; st > 0; st >>= 1) {
        if (tid < st) red[tid] += red[tid + st];
        __syncthreads();
    }
    const float inv = 1.0f / red[0];
#pragma unroll
    for (int i = 0; i < 4; ++i)
        weights[(long)b * SS + tid + 256 * i] = e[i] * inv;
}

// ---------------------------------------------------------------------------
// Pass 2 (L2-bound): partial[b,sch,d] = sum_{s in chunk} w[b,s]*emb[b,s,d]
// grid = (5 dchunks, 8 schunks, 32 b), block = 256 (one d per thread).
// ---------------------------------------------------------------------------
__global__ void pool_partial_kernel(const float* __restrict__ emb,
                                    const float* __restrict__ weights,
                                    float* __restrict__ partial) {
    const int d   = blockIdx.x * 256 + threadIdx.x;
    const int sch = blockIdx.y;
    const int b   = blockIdx.z;

    const float* eb = emb + ((long)b * SS + sch * 128) * DD + d;
    const float* wb = weights + (long)b * SS + sch * 128;

    float acc = 0.0f;
#pragma unroll 4
    for (int s = 0; s < 128; ++s)
        acc += wb[s] * eb[(long)s * DD];

    partial[((long)b * 8 + sch) * DD + d] = acc;
}

__global__ void pool_reduce_kernel(const float* __restrict__ partial,
                                   float* __restrict__ p) {
    const int idx = blockIdx.x * 256 + threadIdx.x;   // 0..B*DD-1
    const int b = idx / DD, d = idx % DD;
    float s = 0.0f;
#pragma unroll
    for (int j = 0; j < 8; ++j)
        s += partial[((long)b * 8 + j) * DD + d];
    p[idx] = s;
}

// ---------------------------------------------------------------------------
// LayerNorm + ReLU. grid=32, block=256 (5 elems/thread over DOUT=1280).
// ---------------------------------------------------------------------------
__global__ void ln_relu_kernel(const float* __restrict__ h,
                               const float* __restrict__ gamma,
                               const float* __restrict__ beta,
                               float* __restrict__ out) {
    __shared__ float r1[256];
    __shared__ float r2[256];
    const int b = blockIdx.x, tid = threadIdx.x;
    const float* hb = h + (long)b * DOUT;

    float v[5];
    float s = 0.0f, s2 = 0.0f;
#pragma unroll
    for (int i = 0; i < 5; ++i) {
        v[i] = hb[tid + 256 * i];
        s += v[i];
        s2 += v[i] * v[i];
    }
    r1[tid] = s; r2[tid] = s2; __syncthreads();
    for (int st = 128; st > 0; st >>= 1) {
        if (tid < st) { r1[tid] += r1[tid + st]; r2[tid] += r2[tid + st]; }
        __syncthreads();
    }
    const float mu  = r1[0] * (1.0f / DOUT);
    const float var = r2[0] * (1.0f / DOUT) - mu * mu;
    const float rs  = rsqrtf(var + LN_EPS);
#pragma unroll
    for (int i = 0; i < 5; ++i) {
        const int d = tid + 256 * i;
        const float y = (v[i] - mu) * rs * gamma[d] + beta[d];
        out[(long)b * DOUT + d] = fmaxf(y, 0.0f);
    }
}

// ---------------------------------------------------------------------------
extern "C" void kernel_launch(void* const* d_in, const int* in_sizes, int n_in,
                              void* d_out, int out_size, void* d_ws, size_t ws_size,
                              hipStream_t stream) {
    const float* emb   = (const float*)d_in[0];
    const int*   mask  = (const int*)  d_in[1];
    const float* Wq    = (const float*)d_in[2];
    const float* bq    = (const float*)d_in[3];
    const float* Wk    = (const float*)d_in[4];
    const float* bk    = (const float*)d_in[5];
    const float* Wv    = (const float*)d_in[6];
    const float* bv    = (const float*)d_in[7];
    const float* Wo    = (const float*)d_in[8];
    const float* bo    = (const float*)d_in[9];
    const float* gamma = (const float*)d_in[10];
    const float* beta  = (const float*)d_in[11];
    float* out = (float*)d_out;

    // workspace layout (floats)
    float* ws      = (float*)d_ws;
    float* q0      = ws;                       // 32*1280
    float* kq      = q0 + BB * DD;             // 32*1280
    float* qbk     = kq + BB * DD;             // 32
    float* scores  = qbk + BB;                 // 32*1024
    float* weights = scores + BB * SS;         // 32*1024
    float* partial = weights + BB * SS;        // 32*8*1280
    float* p       = partial + BB * 8 * DD;    // 32*1280
    float* ctx     = p + BB * DD;              // 32*1280
    float* h       = ctx + BB * DD;            // 32*1280

    const float scale = 1.0f / sqrtf((float)DD);
    const dim3 gtile(DD / 16, 2);              // 80 x 2 tiles, 1 wave each

    // q0 = emb[:,0,:] @ Wq^T + bq
    gemm32_wmma_kernel<1><<<gtile, 32, 0, stream>>>(
        emb, (long)SS * DD, Wq, DD, bq, q0, DD, DD, 1.0f);

    // qbk[b] = scale * (q0[b] . bk)
    qdot_kernel<<<BB, 256, 0, stream>>>(q0, bk, qbk, scale);

    // kq = scale * (q0 @ Wk)
    gemm32_wmma_kernel<0><<<gtile, 32, 0, stream>>>(
        q0, DD, Wk, DD, nullptr, kq, DD, DD, scale);

    // pass 1: masked scores
    scores_kernel<<<(BB * SS) / 8, 256, 0, stream>>>(emb, kq, qbk, mask, scores);

    // softmax
    softmax_kernel<<<BB, 256, 0, stream>>>(scores, weights);

    // pass 2: weighted pooling of embeddings
    pool_partial_kernel<<<dim3(DD / 256, 8, BB), 256, 0, stream>>>(emb, weights, partial);
    pool_reduce_kernel<<<(BB * DD) / 256, 256, 0, stream>>>(partial, p);

    // ctx = p @ Wv^T + bv
    gemm32_wmma_kernel<1><<<gtile, 32, 0, stream>>>(p, DD, Wv, DD, bv, ctx, DD, DD, 1.0f);

    // h = ctx @ Wo^T + bo
    gemm32_wmma_kernel<1><<<gtile, 32, 0, stream>>>(ctx, DD, Wo, DD, bo, h, DD, DD, 1.0f);

    // LayerNorm + ReLU
    ln_relu_kernel<<<BB, 256, 0, stream>>>(h, gamma, beta, out);
}